// Sender_61083024884022
// MI455X (gfx1250) — compile-verified
//
#include <hip/hip_runtime.h>
#include <hip/hip_bf16.h>
#include <math.h>

#define HH   2
#define CC   128
#define HCC  256     // H*C
#define HIDN 256
#define NEG_SLOPE 0.2f
#define CAP  2048    // max compacted edges (expected ~8 for random dst)

typedef __attribute__((ext_vector_type(2))) float v2f;
typedef __attribute__((ext_vector_type(8))) float v8f;

struct EdgeRec { int eid; int src; float a0; float a1; };

// ---------------- ws layout ----------------
// [0,8)      u64 argmax key
// [8,12)     int match count
// [16,1040)  te[256] (float)
// [2048,...) EdgeRec[CAP]

__global__ void k_init(unsigned long long* key, int* cnt) {
    *key = 0ull;
    *cnt = 0;
}

// argmax(labels): max value, smallest index on ties
__global__ void k_argmax(const int* __restrict__ labels, int n,
                         unsigned long long* key) {
    int i = blockIdx.x * blockDim.x + threadIdx.x;
    if (i >= n) return;
    unsigned hv = (unsigned)labels[i] ^ 0x80000000u;   // order-preserving for signed
    unsigned long long k = ((unsigned long long)hv << 32) |
                           (unsigned long long)(0xFFFFFFFFu - (unsigned)i);
    atomicMax(key, k);
}

// scan edges, keep those with dst == target, compute per-head attention logits
__global__ void k_scan(const float* __restrict__ x, const int* __restrict__ ei,
                       const float* __restrict__ eattr, int E,
                       const float* __restrict__ Wl, const float* __restrict__ bl,
                       const float* __restrict__ Wr, const float* __restrict__ br,
                       const float* __restrict__ We, const float* __restrict__ att,
                       const unsigned long long* __restrict__ key,
                       int* cnt, EdgeRec* recs) {
    int e = blockIdx.x * blockDim.x + threadIdx.x;
    if (e >= E) return;
    int target = (int)(0xFFFFFFFFu - (unsigned)((*key) & 0xFFFFFFFFull));
    if (ei[E + e] != target) return;                   // dst row of edge_index

    int s = ei[e];                                     // src row
    float xs0 = x[2 * s],      xs1 = x[2 * s + 1];
    float xt0 = x[2 * target], xt1 = x[2 * target + 1];
    float ea  = eattr[e];
    float a0 = 0.f, a1 = 0.f;
    for (int hc = 0; hc < HCC; ++hc) {
        float m = xs0 * Wl[hc] + xs1 * Wl[HCC + hc] + bl[hc]
                + xt0 * Wr[hc] + xt1 * Wr[HCC + hc] + br[hc]
                + ea  * We[hc];
        float sv = (m > 0.f) ? m : NEG_SLOPE * m;      // leaky_relu
        if (hc < CC) a0 += sv * att[hc];               // head 0
        else         a1 += sv * att[hc];               // head 1 (att flat [H*C])
    }
    int idx = atomicAdd(cnt, 1);
    if (idx < CAP) {
        recs[idx].eid = e; recs[idx].src = s;
        recs[idx].a0 = a0; recs[idx].a1 = a1;
    }
}

// softmax over matched edges + weighted aggregation of xl[src]; one block of 256
__global__ void k_aggregate(const float* __restrict__ x,
                            const float* __restrict__ Wl, const float* __restrict__ bl,
                            const float* __restrict__ bo,
                            const int* __restrict__ cntp,
                            EdgeRec* recs, float* te) {
    int cnt = *cntp; if (cnt > CAP) cnt = CAP;
    // deterministic order: sort compacted list by edge id (tiny list)
    if (threadIdx.x == 0) {
        for (int i = 1; i < cnt; ++i) {
            EdgeRec r = recs[i];
            int j = i - 1;
            while (j >= 0 && recs[j].eid > r.eid) { recs[j + 1] = recs[j]; --j; }
            recs[j + 1] = r;
        }
    }
    __syncthreads();

    float amax0 = -INFINITY, amax1 = -INFINITY;
    for (int i = 0; i < cnt; ++i) {
        amax0 = fmaxf(amax0, recs[i].a0);
        amax1 = fmaxf(amax1, recs[i].a1);
    }
    if (cnt == 0) { amax0 = 0.f; amax1 = 0.f; }        // isfinite fixup in reference
    float den0 = 0.f, den1 = 0.f;
    for (int i = 0; i < cnt; ++i) {
        den0 += expf(recs[i].a0 - amax0);
        den1 += expf(recs[i].a1 - amax1);
    }

    int hc = threadIdx.x;
    int h  = hc >> 7;
    float amax = h ? amax1 : amax0;
    float den  = h ? den1  : den0;
    float acc = 0.f;
    for (int i = 0; i < cnt; ++i) {
        int s = recs[i].src;
        float xl = x[2 * s] * Wl[hc] + x[2 * s + 1] * Wl[HCC + hc] + bl[hc];
        float al = h ? recs[i].a1 : recs[i].a0;
        float a  = expf(al - amax) / (den + 1e-16f);
        acc += xl * a;
    }
    float v = acc + bo[hc];
    te[hc] = v > 0.f ? v : 0.f;                        // relu
}

// out[1,256] = te[1,256] @ Wfc[256,256] + bfc, via V_WMMA_F32_16X16X4_F32.
// One wave32. A broadcasts te into all 16 rows (every row of D computes the
// same dot product, so row M=0 is correct and no row masking / divergent
// loads are needed). ISA layouts (§7.12.2):
//   A 16x4 f32: lanes 0-15 -> K={0,1} in v[0],v[1]; lanes 16-31 -> K={2,3}
//   B 4x16 f32: rows striped across lanes within a VGPR (mirrors A's K split)
//   D 16x16 f32: VGPR0 lanes 0-15 = row M=0, N = lane
__global__ void k_fc_wmma(const float* __restrict__ te,
                          const float* __restrict__ Wfc,
                          const float* __restrict__ bfc,
                          float* __restrict__ out) {
    int lane = threadIdx.x;
    int m  = lane & 15;
    int kh = (lane >> 4) * 2;      // K sub-offset: 0 for lanes 0-15, 2 for 16-31
    float outv[16];
    for (int nt = 0; nt < 16; ++nt) {
        v8f c = {};
        for (int kt = 0; kt < 64; ++kt) {
            int k0 = kt * 4;
            v2f a, b;
            a.x = te[k0 + kh];                         // broadcast: A[m][k]=te[k] all m
            a.y = te[k0 + kh + 1];
            b.x = Wfc[(k0 + kh)     * HIDN + nt * 16 + m];
            b.y = Wfc[(k0 + kh + 1) * HIDN + nt * 16 + m];
            c = __builtin_amdgcn_wmma_f32_16x16x4_f32(
                    false, a, false, b, (short)0, c, false, false);
        }
        outv[nt] = c[0];           // D row M=0 lives in VGPR0, lanes 0-15
    }
    if (lane < 16)
        for (int nt = 0; nt < 16; ++nt)
            out[nt * 16 + lane] = outv[nt] + bfc[nt * 16 + lane];
}

extern "C" void kernel_launch(void* const* d_in, const int* in_sizes, int n_in,
                              void* d_out, int out_size, void* d_ws, size_t ws_size,
                              hipStream_t stream) {
    const float* x      = (const float*)d_in[0];
    const int*   ei     = (const int*)d_in[1];
    const float* eattr  = (const float*)d_in[2];
    const int*   labels = (const int*)d_in[3];
    // conv1 params (d_in[4..10]) are dead code in the reference
    const float* Wl2  = (const float*)d_in[11];
    const float* bl2  = (const float*)d_in[12];
    const float* Wr2  = (const float*)d_in[13];
    const float* br2  = (const float*)d_in[14];
    const float* We2  = (const float*)d_in[15];
    const float* att2 = (const float*)d_in[16];
    const float* bo2  = (const float*)d_in[17];
    const float* Wfc  = (const float*)d_in[18];
    const float* bfc  = (const float*)d_in[19];
    float* out = (float*)d_out;

    int n = in_sizes[3];           // N (labels)
    int E = in_sizes[2];           // E (edge_attr, edge_dim=1)

    char* ws = (char*)d_ws;
    unsigned long long* key = (unsigned long long*)ws;
    int*     cnt  = (int*)(ws + 8);
    float*   te   = (float*)(ws + 16);
    EdgeRec* recs = (EdgeRec*)(ws + 2048);

    k_init<<<1, 1, 0, stream>>>(key, cnt);
    k_argmax<<<(n + 255) / 256, 256, 0, stream>>>(labels, n, key);
    k_scan<<<(E + 255) / 256, 256, 0, stream>>>(x, ei, eattr, E,
                                                Wl2, bl2, Wr2, br2, We2, att2,
                                                key, cnt, recs);
    k_aggregate<<<1, 256, 0, stream>>>(x, Wl2, bl2, bo2, cnt, recs, te);
    k_fc_wmma<<<1, 32, 0, stream>>>(te, Wfc, bfc, out);
}